// LIFSpike_8933531976423
// MI455X (gfx1250) — compile-verified
//
#include <hip/hip_runtime.h>
#include <stdint.h>

// ---------------------------------------------------------------------------
// LIF spiking-neuron scan, MI455X (gfx1250).
//
// x: [64, 8192, 100] fp32, out: spikes same shape. Each (b,n) row is an
// independent sequential recurrence over its 100 *contiguous* timesteps.
// Memory-bound: ~420 MB total traffic -> ~18 us floor @ 23.3 TB/s.
//
// Strategy: lane-per-row; stage tiles of 64 consecutive rows (25,600 B
// contiguous in global memory) into LDS with gfx1250 async copies
// (global_load_async_to_lds_b128, GVS addressing mode: SGPR base + one
// shared VGPR lane offset + 24-bit immediate per chunk -> zero per-load
// address VALU), double-buffered on ASYNCcnt so the copy engine streams
// HBM while the VALU runs the recurrence out of LDS. NT hints on both the
// read-once input stream and write-once output stream (no reuse anywhere,
// dataset > 192 MB L2).
// ---------------------------------------------------------------------------

typedef float v4f __attribute__((ext_vector_type(4)));

constexpr float TAU    = 0.5f;
constexpr float THRESH = 0.5f;
constexpr float BETA   = 0.9f;
constexpr float A_C    = 0.5f;
constexpr float B_C    = 0.5f;

constexpr int T            = 100;
constexpr int ROWS         = 64 * 8192;              // 524288 (b,n) rows
constexpr int THREADS      = 64;                     // 2 waves (wave32)
constexpr int TILE_ROWS    = THREADS;                // one row per lane
constexpr int TILE_FLOATS  = TILE_ROWS * T;          // 6400 floats
constexpr int TILE_BYTES   = TILE_FLOATS * 4;        // 25600 B (contiguous!)
constexpr int LOADS_PER_LN = TILE_BYTES / 16 / THREADS; // 25 b128 per lane
constexpr int NTILES       = ROWS / TILE_ROWS;       // 8192
constexpr int TILES_PER_BLOCK = 4;
constexpr int BLOCKS       = NTILES / TILES_PER_BLOCK; // 2048

// One async global->LDS b128 transfer per J, GVS mode:
//   global address is the tile base in saddr plus the lane offset tid*16 in vaddr plus J*1024
//   LDS address is the value in vdst (bufBase + tid*16) plus the same J*1024
// Template recursion guarantees the "i" immediate is a compile-time constant.
template <int J>
__device__ __forceinline__ void async_load_chain(unsigned laddr, unsigned voff,
                                                 uint64_t sbase) {
    asm volatile("global_load_async_to_lds_b128 %0, %1, %2 offset:%3 th:TH_LOAD_NT"
                 :: "v"(laddr), "v"(voff), "s"(sbase),
                    "i"(J * THREADS * 16)
                 : "memory");
    if constexpr (J + 1 < LOADS_PER_LN) {
        async_load_chain<J + 1>(laddr, voff, sbase);
    }
}

// Issue one tile (25 coalesced wave-wide b128 async copies -> ASYNCcnt += 25).
__device__ __forceinline__ void async_load_tile(const float* __restrict__ x,
                                                unsigned ldsBufBase, int tile,
                                                unsigned voff) {
    uint64_t sbase = (uint64_t)x + (uint64_t)tile * TILE_BYTES; // uniform -> SGPRs
    async_load_chain<0>(ldsBufBase + voff, voff, sbase);
}

__global__ __launch_bounds__(THREADS)
void lif_scan_kernel(const float* __restrict__ x, float* __restrict__ out) {
    __shared__ __align__(16) float lds[2][TILE_FLOATS]; // 51,200 B/block
                                                        // -> 6 blocks / 320KB WGP

    const int tid        = threadIdx.x;
    const int tile0      = blockIdx.x * TILES_PER_BLOCK;
    const unsigned voff  = (unsigned)tid * 16u;         // lane byte offset

    // Low 32 bits of the flat shared address == LDS byte offset.
    unsigned base[2];
    base[0] = (unsigned)(uintptr_t)(&lds[0][0]);
    base[1] = (unsigned)(uintptr_t)(&lds[1][0]);

    // Prime the double-buffer pipeline.
    async_load_tile(x, base[0], tile0 + 0, voff);
    async_load_tile(x, base[1], tile0 + 1, voff);

#pragma unroll
    for (int i = 0; i < TILES_PER_BLOCK; ++i) {
        const int buf = i & 1;

        // Async loads complete in order per wave: with one stage (25 ops)
        // still pending, draining to <=25 means this buffer's stage is done;
        // the final tile drains to 0.
        if (i + 1 < TILES_PER_BLOCK) {
            asm volatile("s_wait_asynccnt 0x19" ::: "memory"); // <= 25
        } else {
            asm volatile("s_wait_asynccnt 0x0" ::: "memory");  // == 0
        }
        __syncthreads(); // all waves' transfers for this buffer are visible

        // Lane owns row (tile*64 + tid): 400 B in LDS, 16B-aligned rows.
        const v4f* __restrict__ rowLds =
            (const v4f*)(&lds[buf][tid * T]);
        v4f* __restrict__ rowOut =
            (v4f*)(out + ((size_t)(tile0 + i) * TILE_ROWS + tid) * T);

        float mem = 0.0f, w = 0.0f;
#pragma unroll
        for (int tc = 0; tc < T / 4; ++tc) {
            v4f xv = rowLds[tc];   // ds_load_b128
            v4f sv;
#pragma unroll
            for (int k = 0; k < 4; ++k) {
                float xt = xv[k];
                // Mirror reference op order exactly:
                mem = mem * TAU + xt - w;
                float spike = ((mem - THRESH) > 0.0f) ? 1.0f : 0.0f;
                w   = BETA * w + (1.0f - BETA) * (A_C * mem + B_C * spike);
                mem = mem - spike * THRESH; // soft reset
                sv[k] = spike;
            }
            // Output is write-once: keep it out of near caches.
            __builtin_nontemporal_store(sv, &rowOut[tc]);
        }

        __syncthreads(); // all waves done reading this buffer (WAR)

        // Prefetch tile i+2 into the buffer we just drained.
        if (i + 2 < TILES_PER_BLOCK) {
            async_load_tile(x, base[buf], tile0 + i + 2, voff);
        }
    }
}

extern "C" void kernel_launch(void* const* d_in, const int* in_sizes, int n_in,
                              void* d_out, int out_size, void* d_ws,
                              size_t ws_size, hipStream_t stream) {
    const float* x = (const float*)d_in[0];
    float* out     = (float*)d_out;
    (void)in_sizes; (void)n_in; (void)out_size; (void)d_ws; (void)ws_size;
    lif_scan_kernel<<<BLOCKS, THREADS, 0, stream>>>(x, out);
}